// CosScorer_27049704030489
// MI455X (gfx1250) — compile-verified
//
#include <hip/hip_runtime.h>
#include <math.h>

typedef __attribute__((ext_vector_type(2))) float v2f;
typedef __attribute__((ext_vector_type(8))) float v8f;
typedef __attribute__((ext_vector_type(4))) int   v4i;

#define BATCH          8
#define T_DIM          8192
#define D_DIM          256
#define S_DIM          200
#define NTILES         13      // 13*16 = 208 >= 200, last tile masked at store
#define ROWS_PER_BLOCK 128     // 8 waves x 16 rows
#define KCHUNK         16
#define NCHUNK         (D_DIM / KCHUNK)   // 16
#define LDS_STRIDE     20      // 16 + 4 pad -> conflict-free ds_load_b64 fragments
#define EPS            1e-8f

#if defined(__gfx1250__) && __has_builtin(__builtin_amdgcn_global_load_async_to_lds_b128)
#define HAVE_ASYNC_LDS 1
#endif

// 16-byte global -> LDS copy: async DMA on CDNA5, sync fallback otherwise.
__device__ __forceinline__ void cp16_g2l(const float* __restrict__ g, float* l) {
#ifdef HAVE_ASYNC_LDS
    __builtin_amdgcn_global_load_async_to_lds_b128((v4i*)g, (v4i*)l,
                                                   /*offset=*/0, /*cpol=*/0);
#else
    *(float4*)l = *(const float4*)g;
#endif
}

__device__ __forceinline__ void wait_async0() {
#ifdef HAVE_ASYNC_LDS
#if __has_builtin(__builtin_amdgcn_s_wait_asynccnt)
    __builtin_amdgcn_s_wait_asynccnt(0);
#else
    asm volatile("s_wait_asynccnt 0" ::: "memory");
#endif
#endif
}

// ---------------------------------------------------------------------------
// Prepass: ny[b*200+s] = ||spk[b,s,:]||  (reads only 1.6 MB, negligible)
// ---------------------------------------------------------------------------
__global__ __launch_bounds__(256) void spk_norm_kernel(const float* __restrict__ spk,
                                                       float* __restrict__ ny) {
    const int wave = threadIdx.x >> 5;
    const int lane = threadIdx.x & 31;
    const int row  = blockIdx.x * 8 + wave;          // [0, 1600)
    if (row >= BATCH * S_DIM) return;
    const float* p = spk + (size_t)row * D_DIM + lane * 8;
    float4 v0 = *(const float4*)(p);
    float4 v1 = *(const float4*)(p + 4);
    float s = v0.x*v0.x + v0.y*v0.y + v0.z*v0.z + v0.w*v0.w
            + v1.x*v1.x + v1.y*v1.y + v1.z*v1.z + v1.w*v1.w;
    #pragma unroll
    for (int off = 16; off >= 1; off >>= 1) s += __shfl_xor(s, off, 32);
    if (lane == 0) ny[row] = sqrtf(s);
}

// ---------------------------------------------------------------------------
// Main kernel: one block = 128 rows of one batch x all 200 spk columns.
// Double-buffered async global->LDS staging overlapped with f32 WMMA.
// ---------------------------------------------------------------------------
__global__ __launch_bounds__(256) void cos_wmma_kernel(const float* __restrict__ xs,
                                                       const float* __restrict__ spk,
                                                       const float* __restrict__ ny,
                                                       float* __restrict__ out) {
    __shared__ float sSpk[2][208 * LDS_STRIDE];           // 2 x 16,640 B
    __shared__ float sX[2][ROWS_PER_BLOCK * LDS_STRIDE];  // 2 x 10,240 B

    const int tid  = threadIdx.x;
    const int wave = tid >> 5;
    const int lane = tid & 31;
    const int m    = lane & 15;   // row (A) / col (B) within 16-tile
    const int hi   = lane >> 4;   // selects K pair {2hi, 2hi+1}

    const int blocksPerBatch = T_DIM / ROWS_PER_BLOCK;  // 64
    const int b       = blockIdx.x / blocksPerBatch;
    const int rowBase = (blockIdx.x % blocksPerBatch) * ROWS_PER_BLOCK;

    const float* xb = xs  + (size_t)b * T_DIM * D_DIM + (size_t)rowBase * D_DIM;
    const float* sb = spk + (size_t)b * S_DIM * D_DIM;

    v8f acc[NTILES];
    #pragma unroll
    for (int t = 0; t < NTILES; ++t) acc[t] = (v8f){0.f,0.f,0.f,0.f,0.f,0.f,0.f,0.f};

    float sq = 0.f;                       // partial ||x||^2 for row myRow
    const int myRow = wave * 16 + m;

    // ---- stage chunk `kb` into buffer `buf` (issue only; completion via asynccnt)
    auto stage = [&](int kb, int buf) {
        // spk chunk: 200 rows x 16 floats = 800 x B128
        for (int i = tid; i < S_DIM * (KCHUNK / 4); i += 256) {
            const int r  = i >> 2;
            const int c4 = (i & 3) * 4;
            cp16_g2l(sb + (size_t)r * D_DIM + kb + c4, &sSpk[buf][r * LDS_STRIDE + c4]);
        }
        // X chunk: 128 rows x 16 floats = 512 x B128
        for (int i = tid; i < ROWS_PER_BLOCK * (KCHUNK / 4); i += 256) {
            const int r  = i >> 2;
            const int c4 = (i & 3) * 4;
            cp16_g2l(xb + (size_t)r * D_DIM + kb + c4, &sX[buf][r * LDS_STRIDE + c4]);
        }
    };

    stage(0, 0);
    wait_async0();
    __syncthreads();

    for (int c = 0; c < NCHUNK; ++c) {
        const int p = c & 1;
        if (c + 1 < NCHUNK) stage((c + 1) * KCHUNK, 1 - p);   // overlap with WMMA below

        #pragma unroll
        for (int kk = 0; kk < KCHUNK; kk += 4) {
            // A fragment: X[myRow][kb+kk+2hi .. +1]
            v2f a = *(const v2f*)&sX[p][myRow * LDS_STRIDE + kk + 2 * hi];
            sq += a.x * a.x + a.y * a.y;
            #pragma unroll
            for (int t = 0; t < NTILES; ++t) {
                // B fragment: spk[t*16+m][kb+kk+2hi .. +1]  (B = spk^T)
                v2f bf = *(const v2f*)&sSpk[p][(t * 16 + m) * LDS_STRIDE + kk + 2 * hi];
                acc[t] = __builtin_amdgcn_wmma_f32_16x16x4_f32(
                    /*neg_a=*/false, a, /*neg_b=*/false, bf,
                    /*c_mod=*/(short)0, acc[t],
                    /*reuse_a=*/false, /*reuse_b=*/false);
            }
        }

        if (c + 1 < NCHUNK) {
            wait_async0();      // this wave's async writes for buffer 1-p done
            __syncthreads();    // everyone's writes visible; reads of buf p done
        }
    }

    // ||x||^2: lane L holds k===2hi,2hi+1 (mod 4); partner lane L^16 holds rest
    float nx2 = sq + __shfl_xor(sq, 16, 32);
    // redistribute to C-layout rows: VGPR r holds row (r + 8*hi)
    float nxr[8];
    #pragma unroll
    for (int r = 0; r < 8; ++r)
        nxr[r] = sqrtf(__shfl(nx2, r + 8 * hi, 32));

    float* ob = out + (size_t)b * T_DIM * S_DIM;
    #pragma unroll
    for (int t = 0; t < NTILES; ++t) {
        const int n = t * 16 + m;
        if (n < S_DIM) {
            const float nyv = ny[b * S_DIM + n];
            #pragma unroll
            for (int r = 0; r < 8; ++r) {
                const int row = rowBase + wave * 16 + r + 8 * hi;
                ob[(size_t)row * S_DIM + n] = acc[t][r] / fmaxf(nxr[r] * nyv, EPS);
            }
        }
    }
}

extern "C" void kernel_launch(void* const* d_in, const int* in_sizes, int n_in,
                              void* d_out, int out_size, void* d_ws, size_t ws_size,
                              hipStream_t stream) {
    const float* xs  = (const float*)d_in[0];   // [8, 8192, 256]
    const float* spk = (const float*)d_in[1];   // [8, 200, 256]
    float*       ny  = (float*)d_ws;            // [8*200] column norms
    float*       out = (float*)d_out;           // [8, 8192, 200]

    spk_norm_kernel<<<(BATCH * S_DIM + 7) / 8, 256, 0, stream>>>(spk, ny);

    const int grid = BATCH * (T_DIM / ROWS_PER_BLOCK);  // 512 blocks
    cos_wmma_kernel<<<grid, 256, 0, stream>>>(xs, spk, ny, out);
}